// WiouWbceLoss_9491877724680
// MI455X (gfx1250) — compile-verified
//
#include <hip/hip_runtime.h>
#include <math.h>

typedef __attribute__((ext_vector_type(2))) float v2f;
typedef __attribute__((ext_vector_type(8))) float v8f;

#define BN 32
#define HN 384
#define WN 384
#define KR 15              // box radius (kernel 31)
#define STRIP_H 16
#define STRIP_W 96
#define TROWS 48           // 46 halo rows padded to 48 (K multiple of 4)
#define TCOLS 128          // 126 halo cols padded to 128
#define TSTR 129           // Tb row stride (floats): avoids 16-way bank conflicts
#define HSTR 108           // Hb row stride: conflict-free for both store/load patterns
#define NTHREADS 256
#define INV_AREA (1.0f / 961.0f)

__global__ void wloss_zero_acc(float* acc) {
    int i = threadIdx.x;
    if (i < BN * 4) acc[i] = 0.0f;
}

__global__ __launch_bounds__(NTHREADS)
void wloss_main(const float* __restrict__ x, const float* __restrict__ t,
                float* __restrict__ acc)
{
    __shared__ float Tb[TROWS * TSTR];   // target halo tile
    __shared__ float Hb[TROWS * HSTR];   // horizontal box sums
    __shared__ float sacc[4];

    const int tid  = threadIdx.x;
    const int wave = tid >> 5;
    const int lane = tid & 31;
    const int lo16 = lane & 15;
    const int hi   = lane >> 4;          // which half-wave (K split for 16x16x4)

    // block -> (image, row strip, col strip)
    int bid = blockIdx.x;
    const int cs  = bid & 3;  bid >>= 2;            // 4 col strips
    const int rs  = bid % 24; bid /= 24;            // 24 row strips
    const int img = bid;                            // 0..31
    const int r0  = rs * STRIP_H;
    const int n0  = cs * STRIP_W;
    const float* timg = t + (size_t)img * HN * WN;
    const float* ximg = x + (size_t)img * HN * WN;

    if (tid < 4) sacc[tid] = 0.0f;

    // ---- phase 0: load zero-padded target halo (48 x 128) into LDS ----
    for (int i = tid; i < TROWS * TCOLS; i += NTHREADS) {
        int tr = i >> 7, tc = i & 127;
        int gr = r0 - KR + tr, gc = n0 - KR + tc;
        float v = 0.0f;
        if (gr >= 0 && gr < HN && gc >= 0 && gc < WN) v = timg[gr * WN + gc];
        Tb[tr * TSTR + tc] = v;
    }
    __syncthreads();

    // ---- phase 1: horizontal 31-tap box sum as A(16x48) x Band(48x16) ----
    // 18 tiles (3 row-groups x 6 col-tiles), wave-strided (wave-uniform branches)
    for (int tile = wave; tile < 18; tile += 8) {
        const int g  = tile / 6;
        const int c0 = (tile % 6) * 16;
        v8f d = {};
        for (int j0 = 0; j0 < 48; j0 += 4) {
            const int ja = j0 + 2 * hi;
            v2f a, b;
            a[0] = Tb[(g * 16 + lo16) * TSTR + c0 + ja];
            a[1] = Tb[(g * 16 + lo16) * TSTR + c0 + ja + 1];
            b[0] = (ja     >= lo16 && ja     <= lo16 + 30) ? 1.0f : 0.0f;
            b[1] = (ja + 1 >= lo16 && ja + 1 <= lo16 + 30) ? 1.0f : 0.0f;
            d = __builtin_amdgcn_wmma_f32_16x16x4_f32(false, a, false, b,
                                                      (short)0, d, false, false);
        }
        for (int v = 0; v < 8; ++v)
            Hb[(g * 16 + v + 8 * hi) * HSTR + c0 + lo16] = d[v];
    }
    __syncthreads();

    // ---- phase 2: vertical 31-tap sum (Band as A, Hb as B) + fused loss ----
    float s_w = 0.0f, s_wb = 0.0f, s_it = 0.0f, s_un = 0.0f;
    for (int ct = wave; ct < 6; ct += 8) {
        const int c0 = ct * 16;
        v8f d = {};
        for (int j0 = 0; j0 < 48; j0 += 4) {
            const int ja = j0 + 2 * hi;
            v2f a, b;
            a[0] = (ja     >= lo16 && ja     <= lo16 + 30) ? 1.0f : 0.0f;
            a[1] = (ja + 1 >= lo16 && ja + 1 <= lo16 + 30) ? 1.0f : 0.0f;
            b[0] = Hb[(ja    ) * HSTR + c0 + lo16];
            b[1] = Hb[(ja + 1) * HSTR + c0 + lo16];
            d = __builtin_amdgcn_wmma_f32_16x16x4_f32(false, a, false, b,
                                                      (short)0, d, false, false);
        }
        for (int v = 0; v < 8; ++v) {
            const int m = v + 8 * hi;          // output row in strip
            const int n = c0 + lo16;           // output col in strip
            const float avg  = d[v] * INV_AREA;
            const float tv   = Tb[(KR + m) * TSTR + KR + n];
            const float xv   = ximg[(r0 + m) * WN + n0 + n];
            const float weit = 1.0f + 5.0f * fabsf(avg - tv);
            const float p    = 1.0f / (1.0f + expf(-xv));
            const float bce  = fmaxf(xv, 0.0f) - xv * tv + log1pf(expf(-fabsf(xv)));
            s_w  += weit;
            s_wb += weit * bce;
            s_it += p * tv * weit;
            s_un += (p + tv) * weit;
        }
    }

    // ---- phase 3: reduce (wave32 shuffle -> LDS atomics -> global atomics) ----
    for (int off = 16; off > 0; off >>= 1) {
        s_w  += __shfl_down(s_w,  off, 32);
        s_wb += __shfl_down(s_wb, off, 32);
        s_it += __shfl_down(s_it, off, 32);
        s_un += __shfl_down(s_un, off, 32);
    }
    if (lane == 0) {
        atomicAdd(&sacc[0], s_w);
        atomicAdd(&sacc[1], s_wb);
        atomicAdd(&sacc[2], s_it);
        atomicAdd(&sacc[3], s_un);
    }
    __syncthreads();
    if (tid < 4) atomicAdd(&acc[img * 4 + tid], sacc[tid]);
}

__global__ void wloss_final(const float* __restrict__ acc, float* __restrict__ out) {
    const int b = threadIdx.x;   // 32 lanes, one image each
    float v = 0.0f;
    if (b < BN) {
        const float sw  = acc[b * 4 + 0];
        const float swb = acc[b * 4 + 1];
        const float sit = acc[b * 4 + 2];
        const float sun = acc[b * 4 + 3];
        const float wbce = swb / sw;
        const float wiou = 1.0f - (sit + 1.0f) / (sun - sit + 1.0f + 1e-7f);
        v = wbce + wiou;
    }
    for (int off = 16; off > 0; off >>= 1) v += __shfl_down(v, off, 32);
    if (b == 0) out[0] = v * (1.0f / BN);
}

extern "C" void kernel_launch(void* const* d_in, const int* in_sizes, int n_in,
                              void* d_out, int out_size, void* d_ws, size_t ws_size,
                              hipStream_t stream) {
    const float* x = (const float*)d_in[0];   // input (logits)
    const float* t = (const float*)d_in[1];   // target
    float* acc = (float*)d_ws;                // 32 images x 4 accumulators
    float* out = (float*)d_out;

    wloss_zero_acc<<<1, 128, 0, stream>>>(acc);
    const int nblocks = BN * (HN / STRIP_H) * (WN / STRIP_W);  // 3072
    wloss_main<<<nblocks, NTHREADS, 0, stream>>>(x, t, acc);
    wloss_final<<<1, 32, 0, stream>>>(acc, out);
}